// LayerRegressionV_43860206027512
// MI455X (gfx1250) — compile-verified
//
#include <hip/hip_runtime.h>

// ---------- types ----------
typedef __attribute__((ext_vector_type(2))) float v2f;
typedef __attribute__((ext_vector_type(8))) float v8f;

static __device__ __forceinline__ v8f wmma_f32(v2f a, v2f b, v8f c) {
    // V_WMMA_F32_16X16X4_F32 : D = A(16x4) * B(4x16) + C(16x16), all f32
    return __builtin_amdgcn_wmma_f32_16x16x4_f32(
        false, a, false, b, (short)0, c, false, false);
}

#define N_WIRES 5
#define DIM 32          // 2^5 amplitudes
#define BATCH 64

// ---------------------------------------------------------------------------
// Kernel 1: precompute each gate's 2x2 complex unitary (8 floats per gate).
// RX: [[c,-is],[-is,c]]  RY: [[c,-s],[s,c]]  RZ: [[e^-ih,0],[0,e^+ih]]
// ---------------------------------------------------------------------------
__global__ void qc_prep_gates(const float* __restrict__ angles,
                              const int* __restrict__ types,
                              float* __restrict__ gates, int n) {
    int g = blockIdx.x * blockDim.x + threadIdx.x;
    if (g >= n) return;
    float half = 0.5f * angles[g];
    float c = __cosf(half), s = __sinf(half);
    int t = types[g];
    float u0, u1, u2, u3, u4, u5, u6, u7; // (u00r,u00i,u01r,u01i,u10r,u10i,u11r,u11i)
    if (t == 0)      { u0=c; u1=0.f;  u2=0.f; u3=-s;  u4=0.f; u5=-s;  u6=c; u7=0.f; }
    else if (t == 1) { u0=c; u1=0.f;  u2=-s;  u3=0.f; u4=s;   u5=0.f; u6=c; u7=0.f; }
    else             { u0=c; u1=-s;   u2=0.f; u3=0.f; u4=0.f; u5=0.f; u6=c; u7=s;  }
    float* o = gates + (size_t)g * 8;
    o[0]=u0; o[1]=u1; o[2]=u2; o[3]=u3; o[4]=u4; o[5]=u5; o[6]=u6; o[7]=u7;
}

// ---------------------------------------------------------------------------
// Apply a 1-qubit gate (stride S pairs) to a 32-amplitude column held in
// registers. Fully unrolled: all indices are compile-time constants.
// ---------------------------------------------------------------------------
template <int S>
static __device__ __forceinline__ void qc_pairs(float* ar, float* ai,
        float u00r, float u00i, float u01r, float u01i,
        float u10r, float u10i, float u11r, float u11i) {
#pragma unroll
    for (int p = 0; p < 16; ++p) {
        const int r0 = ((p & ~(S - 1)) << 1) | (p & (S - 1));
        const int r1 = r0 + S;
        float a0r = ar[r0], a0i = ai[r0], a1r = ar[r1], a1i = ai[r1];
        ar[r0] = u00r*a0r - u00i*a0i + u01r*a1r - u01i*a1i;
        ai[r0] = u00r*a0i + u00i*a0r + u01r*a1i + u01i*a1r;
        ar[r1] = u10r*a0r - u10i*a0i + u11r*a1r - u11i*a1i;
        ai[r1] = u10r*a0i + u10i*a0r + u11r*a1i + u11i*a1r;
    }
}

// ---------------------------------------------------------------------------
// Kernel 2: each workgroup (1 wave of 32) builds the partial 32x32 unitary of
// its contiguous gate chunk. Lane j holds column j in 64 VGPRs.
// Partial layout: [R 32x32 row-major][I 32x32 row-major] = 2048 floats.
// ---------------------------------------------------------------------------
__global__ void qc_chunk_partials(const float* __restrict__ gates,
                                  const int* __restrict__ wires,
                                  float* __restrict__ partials,
                                  int nops, int chunk_size) {
    const int chunk = blockIdx.x;
    const int j = threadIdx.x;
    float ar[DIM], ai[DIM];
#pragma unroll
    for (int r = 0; r < DIM; ++r) { ar[r] = (r == j) ? 1.f : 0.f; ai[r] = 0.f; }

    int g0 = chunk * chunk_size;
    int g1 = g0 + chunk_size; if (g1 > nops) g1 = nops;
    for (int g = g0; g < g1; ++g) {
        const float* u = gates + (size_t)g * 8;   // uniform address -> scalarizable
        float4 ua = *(const float4*)(u);
        float4 ub = *(const float4*)(u + 4);
        int w = wires[g];
        switch (w) {                               // wire 0 = MSB (stride 16)
            case 0: qc_pairs<16>(ar, ai, ua.x,ua.y,ua.z,ua.w, ub.x,ub.y,ub.z,ub.w); break;
            case 1: qc_pairs<8 >(ar, ai, ua.x,ua.y,ua.z,ua.w, ub.x,ub.y,ub.z,ub.w); break;
            case 2: qc_pairs<4 >(ar, ai, ua.x,ua.y,ua.z,ua.w, ub.x,ub.y,ub.z,ub.w); break;
            case 3: qc_pairs<2 >(ar, ai, ua.x,ua.y,ua.z,ua.w, ub.x,ub.y,ub.z,ub.w); break;
            default:qc_pairs<1 >(ar, ai, ua.x,ua.y,ua.z,ua.w, ub.x,ub.y,ub.z,ub.w); break;
        }
    }
    float* outR = partials + (size_t)chunk * 2048;
    float* outI = outR + 1024;
#pragma unroll
    for (int r = 0; r < DIM; ++r) { outR[r * 32 + j] = ar[r]; outI[r * 32 + j] = ai[r]; }
}

// ---------------------------------------------------------------------------
// Kernel 3: tree-combine: out[i] = in[2i+1] * in[2i]  (later chunk x earlier).
// One wave per product; 32x32 complex matmul via V_WMMA_F32_16X16X4_F32.
// Cr = Ar*Br + (-Ai)*Bi ; Ci = Ar*Bi + Ai*Br
// ---------------------------------------------------------------------------
__global__ void qc_combine(const float* __restrict__ in, float* __restrict__ out) {
    const int i = blockIdx.x;
    const float* A = in + (size_t)(2 * i + 1) * 2048;  // later
    const float* B = in + (size_t)(2 * i) * 2048;      // earlier
    float* D = out + (size_t)i * 2048;
    const int l = threadIdx.x, h = l >> 4, m = l & 15;
#pragma unroll
    for (int mt = 0; mt < 2; ++mt) {
#pragma unroll
        for (int nt = 0; nt < 2; ++nt) {
            v8f cr = {}; v8f ci = {};
#pragma unroll
            for (int kb = 0; kb < 32; kb += 4) {
                const int ka = kb + 2 * h;
                v2f a_r = *(const v2f*)(A + (mt * 16 + m) * 32 + ka);
                v2f a_i = *(const v2f*)(A + 1024 + (mt * 16 + m) * 32 + ka);
                v2f a_in; a_in.x = -a_i.x; a_in.y = -a_i.y;
                v2f b_r, b_i;
                const int col = nt * 16 + m;
                b_r.x = B[ka * 32 + col];        b_r.y = B[(ka + 1) * 32 + col];
                b_i.x = B[1024 + ka * 32 + col]; b_i.y = B[1024 + (ka + 1) * 32 + col];
                cr = wmma_f32(a_r,  b_r, cr);
                cr = wmma_f32(a_in, b_i, cr);
                ci = wmma_f32(a_r,  b_i, ci);
                ci = wmma_f32(a_i,  b_r, ci);
            }
#pragma unroll
            for (int v = 0; v < 8; ++v) {
                const int row = mt * 16 + v + 8 * h;
                D[row * 32 + nt * 16 + m] = cr[v];
                D[1024 + row * 32 + nt * 16 + m] = ci[v];
            }
        }
    }
}

// ---------------------------------------------------------------------------
// Kernel 4: initial product states. amp_b[r] = prod_k (bit_k ? -i sin(x/2) : cos(x/2))
// S0 layout: [R 32x64 row=amp col=batch][I 32x64] (ld 64)
// ---------------------------------------------------------------------------
__global__ void qc_init_state(const float* __restrict__ x, float* __restrict__ S0) {
    int t = blockIdx.x * blockDim.x + threadIdx.x;
    if (t >= BATCH * DIM) return;
    int b = t >> 5, r = t & 31;
    float accr = 1.f, acci = 0.f;
#pragma unroll
    for (int k = 0; k < N_WIRES; ++k) {
        float half = 0.5f * x[b * N_WIRES + k];
        float c = __cosf(half), s = __sinf(half);
        int bit = (r >> (4 - k)) & 1;
        float fr = bit ? 0.f : c;
        float fi = bit ? -s  : 0.f;
        float nr = accr * fr - acci * fi;
        float ni = accr * fi + acci * fr;
        accr = nr; acci = ni;
    }
    S0[r * 64 + b] = accr;
    S0[2048 + r * 64 + b] = acci;
}

// ---------------------------------------------------------------------------
// Kernel 5: F(32x64) = U_total(32x32) * S0(32x64), complex, WMMA f32.
// 4 workgroups, each handles one 16-batch column tile and both row tiles.
// ---------------------------------------------------------------------------
__global__ void qc_apply(const float* __restrict__ U, const float* __restrict__ S0,
                         float* __restrict__ F) {
    const int nt = blockIdx.x;                       // 0..3
    const int l = threadIdx.x, h = l >> 4, m = l & 15;
#pragma unroll
    for (int mt = 0; mt < 2; ++mt) {
        v8f cr = {}; v8f ci = {};
#pragma unroll
        for (int kb = 0; kb < 32; kb += 4) {
            const int ka = kb + 2 * h;
            v2f a_r = *(const v2f*)(U + (mt * 16 + m) * 32 + ka);
            v2f a_i = *(const v2f*)(U + 1024 + (mt * 16 + m) * 32 + ka);
            v2f a_in; a_in.x = -a_i.x; a_in.y = -a_i.y;
            v2f b_r, b_i;
            const int col = nt * 16 + m;
            b_r.x = S0[ka * 64 + col];        b_r.y = S0[(ka + 1) * 64 + col];
            b_i.x = S0[2048 + ka * 64 + col]; b_i.y = S0[2048 + (ka + 1) * 64 + col];
            cr = wmma_f32(a_r,  b_r, cr);
            cr = wmma_f32(a_in, b_i, cr);
            ci = wmma_f32(a_r,  b_i, ci);
            ci = wmma_f32(a_i,  b_r, ci);
        }
#pragma unroll
        for (int v = 0; v < 8; ++v) {
            const int row = mt * 16 + v + 8 * h;
            F[row * 64 + nt * 16 + m] = cr[v];
            F[2048 + row * 64 + nt * 16 + m] = ci[v];
        }
    }
}

// ---------------------------------------------------------------------------
// Kernel 6: expectations. <Y_k>_b = 2 * sum_pairs Im(conj(s0) * s1)
// ---------------------------------------------------------------------------
__global__ void qc_expect(const float* __restrict__ F, float* __restrict__ out) {
    int t = threadIdx.x;
    if (t >= BATCH * N_WIRES) return;
    int b = t / N_WIRES, k = t % N_WIRES;
    int S = 1 << (4 - k);
    float sum = 0.f;
#pragma unroll
    for (int p = 0; p < 16; ++p) {
        int r0 = ((p & ~(S - 1)) << 1) | (p & (S - 1));
        int r1 = r0 + S;
        float s0r = F[r0 * 64 + b], s0i = F[2048 + r0 * 64 + b];
        float s1r = F[r1 * 64 + b], s1i = F[2048 + r1 * 64 + b];
        sum += s0r * s1i - s0i * s1r;
    }
    out[b * N_WIRES + k] = 2.f * sum;
}

// ---------------------------------------------------------------------------
extern "C" void kernel_launch(void* const* d_in, const int* in_sizes, int n_in,
                              void* d_out, int out_size, void* d_ws, size_t ws_size,
                              hipStream_t stream) {
    const float* x      = (const float*)d_in[0];
    const float* angles = (const float*)d_in[1];
    const int*   wires  = (const int*)d_in[2];
    const int*   types  = (const int*)d_in[3];
    float* out = (float*)d_out;
    float* ws  = (float*)d_ws;

    const int nops = in_sizes[1];                 // 100000

    // Choose chunk count P (power of two) that fits in workspace (deterministic).
    int P = 512;
    while (P > 1 && ((size_t)nops * 8 + 2 * (size_t)P * 2048 + 8192) * sizeof(float) > ws_size)
        P >>= 1;

    float* gates = ws;                                   // nops*8 floats
    float* bufA  = gates + (size_t)nops * 8;             // P*2048 floats
    float* bufB  = bufA + (size_t)P * 2048;              // P*2048 floats
    float* S0    = bufB + (size_t)P * 2048;              // 4096 floats
    float* F     = S0 + 4096;                            // 4096 floats

    const int cs = (nops + P - 1) / P;

    qc_prep_gates<<<(nops + 255) / 256, 256, 0, stream>>>(angles, types, gates, nops);
    qc_chunk_partials<<<P, 32, 0, stream>>>(gates, wires, bufA, nops, cs);

    float* cur = bufA;
    float* oth = bufB;
    for (int cnt = P; cnt > 1; cnt >>= 1) {
        qc_combine<<<cnt / 2, 32, 0, stream>>>(cur, oth);
        float* tmp = cur; cur = oth; oth = tmp;
    }

    qc_init_state<<<(BATCH * DIM + 255) / 256, 256, 0, stream>>>(x, S0);
    qc_apply<<<4, 32, 0, stream>>>(cur, S0, F);
    qc_expect<<<1, BATCH * N_WIRES, 0, stream>>>(F, out);
    (void)n_in; (void)out_size;
}